// xqdaLoss_68212670595918
// MI455X (gfx1250) — compile-verified
//
#include <hip/hip_runtime.h>
#include <hip/hip_bf16.h>
#include <math.h>

// Problem constants (match reference)
#define NROWS 4096
#define NCOLS 2048
#define NIDS  256
#define NT32  (NCOLS / 32)                   // 64 macro-tile rows/cols (32x32 each)
#define NTRI32 (NT32 * (NT32 + 1) / 2)       // 2080 lower-tri macro tiles
#define NPAIR_TOT 8386560.0f                 // N*(N-1)/2

typedef float v2f __attribute__((ext_vector_type(2)));
typedef float v8f __attribute__((ext_vector_type(8)));

__device__ __forceinline__ v8f wmma4(v2f a, v2f b, v8f c) {
    return __builtin_amdgcn_wmma_f32_16x16x4_f32(false, a, false, b, (short)0, c, false, false);
}

// ---------------- workspace layout (floats unless noted) ----------------
#define WS_S      0
#define WS_T      (WS_S + NIDS * NCOLS)
#define WS_VECP   (WS_T + NCOLS)
#define WS_VECN   (WS_VECP + NCOLS)
#define WS_W      (WS_VECN + NCOLS)
#define WS_RPOS   (WS_W + NROWS)
#define WS_RNEG   (WS_RPOS + NROWS)
#define WS_CNT    (WS_RNEG + NROWS)
#define WS_ACC_B  ((WS_CNT + NIDS) * 4)      // byte offset, 8-aligned
#define WS_BYTES  (WS_ACC_B + 3 * 8)

// ---------------- k1: class histogram ----------------
__global__ void k_hist(const int* __restrict__ tgt, int* __restrict__ counts) {
    int i = blockIdx.x * blockDim.x + threadIdx.x;
    if (i < NROWS) atomicAdd(&counts[tgt[i]], 1);
}

// ---------------- k2: pair coefficients r_pos / r_neg, row weights ----------------
__global__ void k_coeff(const int* __restrict__ tgt, const int* __restrict__ counts,
                        float* __restrict__ w, float* __restrict__ rpos,
                        float* __restrict__ rneg) {
    __shared__ int tl[NROWS];
    for (int j = threadIdx.x; j < NROWS; j += blockDim.x) tl[j] = tgt[j];
    __syncthreads();
    int i = blockIdx.x * blockDim.x + threadIdx.x;
    if (i >= NROWS) return;
    int ti = tl[i];
    int below = 0;
    for (int j = 0; j < i; ++j) below += (tl[j] == ti) ? 1 : 0;
    int n = counts[ti];
    w[i] = (float)n;
    float rp = 2.0f * (float)below - (float)(n - 1);
    rpos[i] = rp;
    rneg[i] = (2.0f * (float)i - (float)(NROWS - 1)) - rp;
}

// ---------------- k3: class sums S, totals T, vec_pos/vec_neg ----------------
__global__ void k_colsum(const float* __restrict__ x, const int* __restrict__ tgt,
                         const float* __restrict__ rpos, const float* __restrict__ rneg,
                         float* __restrict__ S, float* __restrict__ T,
                         float* __restrict__ vecp, float* __restrict__ vecn) {
    int f = blockIdx.x * blockDim.x + threadIdx.x;   // 0..2047, exclusive column owner
    float tt = 0.f, vp = 0.f, vn = 0.f;
    for (int k = 0; k < NROWS; ++k) {
        float v = x[k * NCOLS + f];
        S[tgt[k] * NCOLS + f] += v;
        tt += v;
        vp += rpos[k] * v;
        vn += rneg[k] * v;
    }
    T[f] = tt; vecp[f] = vp; vecn[f] = vn;
}

// ---------------- k4: WMMA fp32 Gram, 32x32 macro tile per wave ----------------
// Each wave accumulates a 2x2 grid of 16x16 WMMA subtiles for G1 = X^T X and
// Gn = X^T diag(n) X (8 x v_wmma_f32_16x16x4_f32 per K=4 step, 1 load : 1 wmma),
// then P = S^T S over K=256. Epilogue forms cov_pos / cov_neg / diff, squares,
// wave32-reduces, and f64-atomically accumulates the three Frobenius sums.
__global__ void __launch_bounds__(256)
k_gram(const float* __restrict__ x, const float* __restrict__ S,
       const float* __restrict__ T, const float* __restrict__ w,
       double* __restrict__ acc) {
    __shared__ float wlds[NROWS];                    // 16 KB of n_{t_k}
    for (int i = threadIdx.x; i < NROWS; i += blockDim.x) wlds[i] = w[i];
    __syncthreads();

    const int lane = threadIdx.x & 31;
    const int wv   = threadIdx.x >> 5;
    const int l    = blockIdx.x * 8 + wv;            // linear lower-tri macro-tile id
    if (l >= NTRI32) return;

    // decode tr >= tc from l = tr*(tr+1)/2 + tc
    int tr = (int)((sqrtf(8.0f * (float)l + 1.0f) - 1.0f) * 0.5f);
    while (tr * (tr + 1) / 2 > l) --tr;
    while ((tr + 1) * (tr + 2) / 2 <= l) ++tr;
    const int tc = l - tr * (tr + 1) / 2;
    const int u0 = tr * 32;                          // M-dim feature base
    const int v0 = tc * 32;                          // N-dim feature base

    const int m  = lane & 15;                        // M/N index held by this lane
    const int kh = (lane >> 4) << 1;                 // K sub-offset: 0 or 2

    v8f g100 = {}, g101 = {}, g110 = {}, g111 = {};
    v8f gn00 = {}, gn01 = {}, gn10 = {}, gn11 = {};
    const float* xa = x + u0 + m;                    // second subtile at +16
    const float* xb = x + v0 + m;

    for (int k0 = 0; k0 < NROWS; k0 += 8) {
        // speculative prefetch of the K-panel 64 rows ahead (dropped past the end)
        __builtin_prefetch(xa + (k0 + 64) * NCOLS, 0, 1);
        __builtin_prefetch(xb + (k0 + 64) * NCOLS, 0, 1);
#pragma unroll
        for (int s = 0; s < 2; ++s) {
            const int ka = k0 + 4 * s + kh;
            const int i0 = ka * NCOLS, i1 = i0 + NCOLS;
            v2f a0, a1, b0, b1, as0, as1;
            a0.x = xa[i0];      a0.y = xa[i1];
            a1.x = xa[i0 + 16]; a1.y = xa[i1 + 16];
            b0.x = xb[i0];      b0.y = xb[i1];
            b1.x = xb[i0 + 16]; b1.y = xb[i1 + 16];
            const float w0 = wlds[ka], w1 = wlds[ka + 1];
            as0.x = a0.x * w0; as0.y = a0.y * w1;
            as1.x = a1.x * w0; as1.y = a1.y * w1;
            g100 = wmma4(a0, b0, g100);  g101 = wmma4(a0, b1, g101);
            g110 = wmma4(a1, b0, g110);  g111 = wmma4(a1, b1, g111);
            gn00 = wmma4(as0, b0, gn00); gn01 = wmma4(as0, b1, gn01);
            gn10 = wmma4(as1, b0, gn10); gn11 = wmma4(as1, b1, gn11);
        }
    }

    v8f p00 = {}, p01 = {}, p10 = {}, p11 = {};
    const float* sa = S + u0 + m;
    const float* sb = S + v0 + m;
#pragma unroll 2
    for (int k0 = 0; k0 < NIDS; k0 += 4) {
        const int ka = k0 + kh;
        const int i0 = ka * NCOLS, i1 = i0 + NCOLS;
        v2f a0, a1, b0, b1;
        a0.x = sa[i0];      a0.y = sa[i1];
        a1.x = sa[i0 + 16]; a1.y = sa[i1 + 16];
        b0.x = sb[i0];      b0.y = sb[i1];
        b1.x = sb[i0 + 16]; b1.y = sb[i1 + 16];
        p00 = wmma4(a0, b0, p00); p01 = wmma4(a0, b1, p01);
        p10 = wmma4(a1, b0, p10); p11 = wmma4(a1, b1, p11);
    }

    // epilogue: subtile (R,C), element (M,N) -> M = q + 8*(lane>=16), N = lane&15
    // diagonal macro tile: off-diag subtiles are mutual transposes -> uniform wt
    const float wt   = (tr == tc) ? 1.0f : 2.0f;
    const int   moff = (lane >> 4) << 3;
    float sp = 0.f, sn = 0.f, sd = 0.f;
#define EPI(G1A, GNA, PA, RR, CC)                                         \
    {                                                                     \
        const float tcol = T[v0 + 16 * (CC) + m];                         \
        const int   mb   = u0 + 16 * (RR) + moff;                         \
        _Pragma("unroll")                                                 \
        for (int q = 0; q < 8; ++q) {                                     \
            float trow = T[mb + q];                                       \
            float cp = GNA[q] - PA[q];                                    \
            float cn = (float)NROWS * G1A[q] - GNA[q] - trow * tcol + PA[q]; \
            float cd = cp - cn;                                           \
            sp += cp * cp; sn += cn * cn; sd += cd * cd;                  \
        }                                                                 \
    }
    EPI(g100, gn00, p00, 0, 0)
    EPI(g101, gn01, p01, 0, 1)
    EPI(g110, gn10, p10, 1, 0)
    EPI(g111, gn11, p11, 1, 1)
#undef EPI
    sp *= wt; sn *= wt; sd *= wt;

#pragma unroll
    for (int off = 16; off > 0; off >>= 1) {
        sp += __shfl_xor(sp, off, 32);
        sn += __shfl_xor(sn, off, 32);
        sd += __shfl_xor(sd, off, 32);
    }
    if (lane == 0) {
        atomicAdd(&acc[0], (double)sp);
        atomicAdd(&acc[1], (double)sn);
        atomicAdd(&acc[2], (double)sd);
    }
}

// ---------------- k5: finalize the 4 scalars ----------------
__global__ void k_final(const float* __restrict__ vecp, const float* __restrict__ vecn,
                        const int* __restrict__ counts, const double* __restrict__ acc,
                        const int* __restrict__ hyper, const int* __restrict__ usemean,
                        const int* __restrict__ use_exp, float* __restrict__ out) {
    __shared__ float redf[256];
    __shared__ int   redi[256];
    const int tid = threadIdx.x;

    int n = counts[tid];                       // 256 classes, 256 threads
    redi[tid] = n * (n - 1) / 2;
    __syncthreads();
    for (int s = 128; s > 0; s >>= 1) {
        if (tid < s) redi[tid] += redi[tid + s];
        __syncthreads();
    }
    const float npos = (float)redi[0];
    const float nneg = NPAIR_TOT - npos;
    __syncthreads();

    float vd = 0.f;
    for (int f = tid; f < NCOLS; f += 256) {
        float d = vecp[f] / npos - vecn[f] / nneg;
        vd += d * d;
    }
    redf[tid] = vd;
    __syncthreads();
    for (int s = 128; s > 0; s >>= 1) {
        if (tid < s) redf[tid] += redf[tid + s];
        __syncthreads();
    }

    if (tid == 0) {
        float norm_i = sqrtf((float)acc[0]);
        float norm_j = sqrtf((float)acc[1]);
        float dist, loss;
        if (use_exp[0]) {
            dist = expf(-sqrtf((float)acc[2]));
            loss = dist;
        } else {
            float mean_norm = sqrtf((float)NCOLS) * sqrtf(redf[0]);
            dist = norm_i - (float)hyper[0] * norm_j - (float)usemean[0] * mean_norm;
            loss = fmaxf(dist, 0.0f);
        }
        out[0] = loss; out[1] = dist; out[2] = norm_i; out[3] = norm_j;
    }
}

extern "C" void kernel_launch(void* const* d_in, const int* in_sizes, int n_in,
                              void* d_out, int out_size, void* d_ws, size_t ws_size,
                              hipStream_t stream) {
    const float* x   = (const float*)d_in[0];
    const int*   tgt = (const int*)d_in[1];
    const int*   hp  = (const int*)d_in[2];
    const int*   um  = (const int*)d_in[3];
    const int*   ue  = (const int*)d_in[4];
    float* out = (float*)d_out;

    float*  Sw    = (float*)d_ws + WS_S;
    float*  Tw    = (float*)d_ws + WS_T;
    float*  vecp  = (float*)d_ws + WS_VECP;
    float*  vecn  = (float*)d_ws + WS_VECN;
    float*  wv    = (float*)d_ws + WS_W;
    float*  rpos  = (float*)d_ws + WS_RPOS;
    float*  rneg  = (float*)d_ws + WS_RNEG;
    int*    cnt   = (int*)((float*)d_ws + WS_CNT);
    double* acc   = (double*)((char*)d_ws + WS_ACC_B);

    hipMemsetAsync(d_ws, 0, WS_BYTES, stream);

    k_hist  <<<NROWS / 256, 256, 0, stream>>>(tgt, cnt);
    k_coeff <<<NROWS / 256, 256, 0, stream>>>(tgt, cnt, wv, rpos, rneg);
    k_colsum<<<NCOLS / 256, 256, 0, stream>>>(x, tgt, rpos, rneg, Sw, Tw, vecp, vecn);
    k_gram  <<<NTRI32 / 8,  256, 0, stream>>>(x, Sw, Tw, wv, acc);
    k_final <<<1,           256, 0, stream>>>(vecp, vecn, cnt, acc, hp, um, ue, out);
}